// OctreeConv2D_15479062134907
// MI455X (gfx1250) — compile-verified
//
#include <hip/hip_runtime.h>

typedef __attribute__((ext_vector_type(2))) float v2f;
typedef __attribute__((ext_vector_type(8))) float v8f;

#define N_NZ   131072
#define N_MASK 131072
#define BATCH  2
#define Hc     512
#define Wc     512
#define CIN    64
#define COUT   64

// ---------------- zero fill ----------------
__global__ void zero_f32(float* __restrict__ p, int n) {
  int i = blockIdx.x * blockDim.x + threadIdx.x;
  if (i < n) p[i] = 0.0f;
}

// ---------------- mask scatter ----------------
__global__ void mask_scatter(const int* __restrict__ mi,
                             const float* __restrict__ mv,
                             float* __restrict__ mask) {
  int i = blockIdx.x * blockDim.x + threadIdx.x;
  if (i < N_MASK) {
    int b = mi[3 * i + 0];
    int y = mi[3 * i + 1];
    int x = mi[3 * i + 2];
    atomicAdd(&mask[(b * Hc + y) * Wc + x], mv[i]);
  }
}

// ---------------- WMMA conv + scatter ----------------
// One wave32 owns a 16-point tile. Per tap: D(16x64) = A(16x64) x B(64x64)
// via chained V_WMMA_F32_16X16X4_F32 (16 K-steps x 4 N-tiles), then scatter.
__launch_bounds__(128)
__global__ void conv_wmma(const float* __restrict__ values,
                          const float* __restrict__ kern,
                          const int*  __restrict__ idx,
                          float* __restrict__ out) {
  __shared__ float ktile[CIN * COUT];   // 16 KB: one tap's 64x64 weights

  const int lane = threadIdx.x & 31;
  const int wave = threadIdx.x >> 5;
  const int tile = blockIdx.x * 4 + wave;
  const int p0   = tile * 16;

  const int row = lane & 15;    // M for A, N for B/C/D
  const int hi  = lane >> 4;    // lane half
  const int kp  = hi * 2;       // K-pair offset within each group of 4

  // A-fragments (ISA 32-bit 16x4 layout): frag j covers K=4j..4j+3
  v2f a[16];
  const float* vp = values + (size_t)(p0 + row) * CIN;
  #pragma unroll
  for (int j = 0; j < 16; ++j) {
    a[j].x = vp[4 * j + kp + 0];
    a[j].y = vp[4 * j + kp + 1];
  }

  // point coordinates live in lanes 0..15 (lane == point-in-tile)
  int ib = 0, iy = 0, ix = 0;
  if (lane < 16) {
    ib = idx[3 * (p0 + lane) + 0];
    iy = idx[3 * (p0 + lane) + 1];
    ix = idx[3 * (p0 + lane) + 2];
  }

  for (int ky = 0; ky < 3; ++ky) {
    for (int kx = 0; kx < 3; ++kx) {
      // stage this tap's 64x64 weight slice into LDS (cooperative, float4)
      __syncthreads();
      {
        const float4* src = (const float4*)(kern + (size_t)((ky * 3 + kx) * CIN) * COUT);
        float4* dst = (float4*)ktile;
        #pragma unroll
        for (int t = threadIdx.x; t < (CIN * COUT) / 4; t += 128) dst[t] = src[t];
      }
      __syncthreads();

      // destination pixel base offset for this tap (valid in lanes 0..15)
      int sy = iy + ky - 1; sy = sy < 0 ? 0 : (sy > Hc - 1 ? Hc - 1 : sy);
      int sx = ix + kx - 1; sx = sx < 0 ? 0 : (sx > Wc - 1 ? Wc - 1 : sx);
      int pixbase = (((ib * Hc) + sy) * Wc + sx) * COUT;

      #pragma unroll
      for (int nt = 0; nt < 4; ++nt) {
        v8f c = {};
        #pragma unroll
        for (int j = 0; j < 16; ++j) {
          // B-fragment 4x16 (lanes 0-15: rows K=4j+0/1; lanes 16-31: rows K=4j+2/3)
          v2f b;
          b.x = ktile[(4 * j + kp + 0) * COUT + nt * 16 + row];
          b.y = ktile[(4 * j + kp + 1) * COUT + nt * 16 + row];
          c = __builtin_amdgcn_wmma_f32_16x16x4_f32(
              /*neg_a=*/false, a[j], /*neg_b=*/false, b,
              /*c_mod=*/(short)0, c, /*reuse_a=*/false, /*reuse_b=*/false);
        }
        // scatter: acc VGPR r -> (M = r + hi*8, N = nt*16 + row)
        #pragma unroll
        for (int r = 0; r < 8; ++r) {
          int m    = r + hi * 8;
          int base = __shfl(pixbase, m, 32);
          atomicAdd(&out[base + nt * 16 + row], c[r]);
        }
      }
    }
  }
}

// ---------------- finalize: out = (out + mask*bias) * mask ----------------
__global__ void finalize(float* __restrict__ out,
                         const float* __restrict__ mask,
                         const float* __restrict__ bias) {
  int g = blockIdx.x * blockDim.x + threadIdx.x;   // one float4 per thread
  int pix = g >> 4;                // 16 float4 per pixel (64 channels)
  int c4  = (g & 15) * 4;
  float m = mask[pix];
  float4 v = ((float4*)out)[g];
  v.x = (v.x + m * bias[c4 + 0]) * m;
  v.y = (v.y + m * bias[c4 + 1]) * m;
  v.z = (v.z + m * bias[c4 + 2]) * m;
  v.w = (v.w + m * bias[c4 + 3]) * m;
  ((float4*)out)[g] = v;
}

extern "C" void kernel_launch(void* const* d_in, const int* in_sizes, int n_in,
                              void* d_out, int out_size, void* d_ws, size_t ws_size,
                              hipStream_t stream) {
  const float* values       = (const float*)d_in[0];
  const float* kern         = (const float*)d_in[1];
  const float* bias         = (const float*)d_in[2];
  const float* mask_values  = (const float*)d_in[3];
  const int*   indices      = (const int*)d_in[4];
  const int*   mask_indices = (const int*)d_in[5];
  float* out     = (float*)d_out;
  float* maskbuf = (float*)d_ws;   // BATCH*Hc*Wc floats = 2 MB

  const int outN  = BATCH * Hc * Wc * COUT;   // 33.5M
  const int maskN = BATCH * Hc * Wc;          // 524288

  zero_f32<<<(outN + 255) / 256, 256, 0, stream>>>(out, outN);
  zero_f32<<<(maskN + 255) / 256, 256, 0, stream>>>(maskbuf, maskN);
  mask_scatter<<<(N_MASK + 255) / 256, 256, 0, stream>>>(mask_indices, mask_values, maskbuf);
  conv_wmma<<<N_NZ / 64, 128, 0, stream>>>(values, kern, indices, out);
  finalize<<<(outN / 4 + 255) / 256, 256, 0, stream>>>(out, maskbuf, bias);
}